// YOLOWrapper_88708254531775
// MI455X (gfx1250) — compile-verified
//
#include <hip/hip_runtime.h>
#include <hip/hip_bf16.h>
#include <math.h>

#define IMG 640
#define RSZ 224
#define NPRED 25200
#define NIMG 32
#define CONF 0.25f

// ---- gfx1250 async global->LDS path detection ----
#if defined(__HIP_DEVICE_COMPILE__)
#if __has_builtin(__builtin_amdgcn_global_load_async_to_lds_b128)
#define HAVE_ASYNC_LDS 1
#endif
#endif

// Builtin signature (from hipcc diagnostic): param0 = v4i AS(1)*, param1 = v4i AS(3)*
typedef int v4i_t __attribute__((vector_size(4 * sizeof(int))));
typedef __attribute__((address_space(1))) v4i_t gv4i_t;
typedef __attribute__((address_space(3))) v4i_t lv4i_t;

__device__ __forceinline__ void wait_async_zero() {
#if defined(__HIP_DEVICE_COMPILE__)
#if __has_builtin(__builtin_amdgcn_s_wait_asynccnt)
    __builtin_amdgcn_s_wait_asynccnt(0);
#else
    asm volatile("s_wait_asynccnt 0" ::: "memory");
#endif
#endif
}

// Kernel 1: per-image masked argmax + box decode + separable-resize index/weight tables.
__global__ void yolo_nms_tables_kernel(const float* __restrict__ pred,
                                       float* __restrict__ has_out,
                                       int* __restrict__ iy0t, int* __restrict__ iy1t,
                                       int* __restrict__ ix0t, int* __restrict__ ix1t,
                                       float* __restrict__ fyt, float* __restrict__ fxt) {
    const int b = blockIdx.x;
    const int tid = threadIdx.x;
    const float* p = pred + (size_t)b * NPRED * 6;

    __shared__ float sval[256];
    __shared__ int   sidx[256];
    __shared__ int   sparams[4]; // y1, x1, h, w

    // masked argmax of obj (nc==1 -> conf==obj; valid == obj > CONF)
    float best = -INFINITY;
    int   bidx = 0x7fffffff;
    for (int i = tid; i < NPRED; i += 256) {
        float obj = p[(size_t)i * 6 + 4];
        if (obj > CONF && obj > best) { best = obj; bidx = i; }  // strict > keeps first index
    }
    sval[tid] = best; sidx[tid] = bidx;
    __syncthreads();
    for (int s = 128; s > 0; s >>= 1) {
        if (tid < s) {
            float v1 = sval[tid], v2 = sval[tid + s];
            int   i1 = sidx[tid], i2 = sidx[tid + s];
            if (v2 > v1 || (v2 == v1 && i2 < i1)) { sval[tid] = v2; sidx[tid] = i2; }
        }
        __syncthreads();
    }

    if (tid == 0) {
        const int has = (sval[0] > -INFINITY) ? 1 : 0;
        int x1 = 0, y1 = 0, w = IMG, h = IMG;
        if (has) {
            const int idx = sidx[0];
            float b0 = p[(size_t)idx * 6 + 0];
            float b1 = p[(size_t)idx * 6 + 1];
            float b2 = p[(size_t)idx * 6 + 2];
            float b3 = p[(size_t)idx * 6 + 3];
            float xa = fminf(fmaxf(b0 - b2 * 0.5f, 0.0f), (float)IMG);
            float ya = fminf(fmaxf(b1 - b3 * 0.5f, 0.0f), (float)IMG);
            float xb = fminf(fmaxf(b0 + b2 * 0.5f, 0.0f), (float)IMG);
            float yb = fminf(fmaxf(b1 + b3 * 0.5f, 0.0f), (float)IMG);
            int bi0 = (int)rintf(xa);   // rintf = round-half-even == jnp.round
            int bi1 = (int)rintf(ya);
            int bi2 = (int)rintf(xb);
            int bi3 = (int)rintf(yb);
            x1 = bi0; y1 = bi1;
            w = max(bi2 - bi0, 1);
            h = max(bi3 - bi1, 1);
        }
        sparams[0] = y1; sparams[1] = x1; sparams[2] = h; sparams[3] = w;
        has_out[b] = (float)has;
    }
    __syncthreads();

    const int y1 = sparams[0], x1 = sparams[1], h = sparams[2], w = sparams[3];
    // bilinear (align_corners=False) tables for both axes, exactly as _resize_weights
    for (int t = tid; t < 2 * RSZ; t += 256) {
        const bool isY = (t < RSZ);
        const int  j   = isY ? t : (t - RSZ);
        const int  s0  = isY ? y1 : x1;
        const int  L   = isY ? h : w;
        const float lf = (float)L;
        const float tt = ((float)j + 0.5f) * (lf / (float)RSZ) - 0.5f;
        const float t0 = floorf(tt);
        const float f  = tt - t0;
        int i0 = min(max((int)t0, 0), L - 1) + s0;
        int i1 = min(max((int)t0 + 1, 0), L - 1) + s0;
        i0 = min(i0, IMG - 1);   // safety clamp (unreachable for this data range)
        i1 = min(i1, IMG - 1);
        const int o = b * RSZ + j;
        if (isY) { iy0t[o] = i0; iy1t[o] = i1; fyt[o] = f; }
        else     { ix0t[o] = i0; ix1t[o] = i1; fxt[o] = f; }
    }
}

// Kernel 2: one block per (image, channel, output row). Async-stage the two
// source rows into LDS (global_load_async_to_lds_b128), then 224 lanes each
// compute one bilinear output pixel (y-combine first, then x: matches Wy@img@Wx^T).
__global__ void yolo_crop_resize_kernel(const float* __restrict__ x,
                                        float* __restrict__ out,
                                        const int* __restrict__ iy0t, const int* __restrict__ iy1t,
                                        const int* __restrict__ ix0t, const int* __restrict__ ix1t,
                                        const float* __restrict__ fyt, const float* __restrict__ fxt) {
    const int bid = blockIdx.x;
    const int j   = bid % RSZ;
    const int bc  = bid / RSZ;          // 0..95 : image*3 + channel

    __shared__ float rows[2 * IMG];     // row iy0 then row iy1 (full 640-wide rows)

    const int b = bc / 3;
    const int o = b * RSZ + j;
    const int iy0 = iy0t[o];
    const int iy1 = iy1t[o];
    const float fy = fyt[o];

    const float* img = x + (size_t)bc * IMG * IMG;

    // 2 rows * 640 f32 = 320 x 16B chunks; rows are 2560B-aligned so b128 is aligned.
    for (int i = threadIdx.x; i < 320; i += 256) {
        const int r  = (i >= 160) ? 1 : 0;
        const int cc = i - r * 160;
        const int yrow = r ? iy1 : iy0;
        const float* g = img + (size_t)yrow * IMG + cc * 4;
        float* l = &rows[r * IMG + cc * 4];
#if defined(HAVE_ASYNC_LDS)
        __builtin_amdgcn_global_load_async_to_lds_b128((gv4i_t*)g, (lv4i_t*)l, 0, 0);
#else
        *(float4*)l = *(const float4*)g;
#endif
    }
#if defined(HAVE_ASYNC_LDS)
    wait_async_zero();
#endif
    __syncthreads();

    const int k = threadIdx.x;
    if (k < RSZ) {
        const int ko = b * RSZ + k;
        const int ix0 = ix0t[ko];
        const int ix1 = ix1t[ko];
        const float fx = fxt[ko];
        const float wy0 = 1.0f - fy;
        const float c0 = wy0 * rows[ix0]       + fy * rows[IMG + ix0];
        const float c1 = wy0 * rows[ix1]       + fy * rows[IMG + ix1];
        const float v  = (1.0f - fx) * c0 + fx * c1;
        out[((size_t)bc * RSZ + j) * RSZ + k] = v;
    }
}

extern "C" void kernel_launch(void* const* d_in, const int* in_sizes, int n_in,
                              void* d_out, int out_size, void* d_ws, size_t ws_size,
                              hipStream_t stream) {
    const float* x    = (const float*)d_in[0];   // [32,3,640,640] f32
    const float* pred = (const float*)d_in[1];   // [32,25200,6]  f32
    float* out = (float*)d_out;                  // images [32,3,224,224] then has_bowl [32,1]
    float* has_out = out + (size_t)NIMG * 3 * RSZ * RSZ;

    int*   iy0t = (int*)d_ws;
    int*   iy1t = iy0t + NIMG * RSZ;
    int*   ix0t = iy1t + NIMG * RSZ;
    int*   ix1t = ix0t + NIMG * RSZ;
    float* fyt  = (float*)(ix1t + NIMG * RSZ);
    float* fxt  = fyt + NIMG * RSZ;

    yolo_nms_tables_kernel<<<NIMG, 256, 0, stream>>>(pred, has_out,
                                                     iy0t, iy1t, ix0t, ix1t, fyt, fxt);
    yolo_crop_resize_kernel<<<NIMG * 3 * RSZ, 256, 0, stream>>>(x, out,
                                                     iy0t, iy1t, ix0t, ix1t, fyt, fxt);
}